// DecoderLinear_for_EffectiveLP_multiclass_33268816675219
// MI455X (gfx1250) — compile-verified
//
#include <hip/hip_runtime.h>

typedef float v2f __attribute__((ext_vector_type(2)));
typedef float v4f __attribute__((ext_vector_type(4)));
typedef float v8f __attribute__((ext_vector_type(8)));

#define N_NODES 4096
#define DIMS    256

// ---------------------------------------------------------------------------
// Phase 1: src/dst projections via V_WMMA_F32_16X16X4_F32.
// One wave (32 lanes) per 16-row tile of x.
//   A (16x4 f32): lane L holds row m=L&15, K pair (L>>4)*2 {+0,+1} in 2 VGPRs.
//   B (4x16 f32): column 0 = w_src chunk, column 1 = w_dst chunk, rest zero.
//   D (16x16 f32): column 0 accumulates src, column 1 accumulates dst.
// B is built ARITHMETICALLY (w*f0 + u*f1 with per-lane f0/f1 in {0,1}) so the
// compiler cannot convert the column select back into divergent loads: the
// loop body is 3x global_load_b64 + FMAs + 1 WMMA, branch-free.
// ---------------------------------------------------------------------------
__global__ __launch_bounds__(32) void proj_wmma(
    const float* __restrict__ x,
    const float* __restrict__ w_src, const float* __restrict__ b_src,
    const float* __restrict__ w_dst, const float* __restrict__ b_dst,
    float* __restrict__ src, float* __restrict__ dst)
{
    const int lane = threadIdx.x;        // 0..31
    const int tile = blockIdx.x;         // 0..255
    const int m    = lane & 15;          // A row / B column this lane touches
    const int kgrp = (lane >> 4) << 1;   // 0 or 2: K sub-pair this lane owns

    // Per-lane blend factors: lane column 0 -> w_src, column 1 -> w_dst,
    // all other columns -> 0. Runtime values, so no dead-load elimination.
    const float f0 = (m == 0) ? 1.0f : 0.0f;
    const float f1 = (m == 1) ? 1.0f : 0.0f;

    const float* xrow = x + (size_t)(tile * 16 + m) * DIMS + kgrp;
    const float* wsp  = w_src + kgrp;    // 8-byte aligned (kgrp even)
    const float* wdp  = w_dst + kgrp;

    v8f acc = {};
    #pragma unroll 4
    for (int kb = 0; kb < DIMS; kb += 4) {
        const v2f a  = *(const v2f*)(xrow + kb);   // A fragment   (b64 load)
        const v2f wv = *(const v2f*)(wsp + kb);    // w_src pair   (b64 load)
        const v2f uv = *(const v2f*)(wdp + kb);    // w_dst pair   (b64 load)

        v2f b;
        b.x = fmaf(wv.x, f0, uv.x * f1);
        b.y = fmaf(wv.y, f0, uv.y * f1);

        // 8 args: (neg_a, A, neg_b, B, c_mod, C, reuse_a, reuse_b)
        acc = __builtin_amdgcn_wmma_f32_16x16x4_f32(
            false, a, false, b, (short)0, acc, false, false);
    }

    // D layout: VGPR r, lanes 0-15 -> D[r][n]; lanes 16-31 -> D[r+8][n]
    const int half = lane >> 4;
    const int base = tile * 16 + half * 8;
    if (m == 0) {
        const float bs = b_src[0];
        #pragma unroll
        for (int r = 0; r < 8; ++r) src[base + r] = acc[r] + bs;
    } else if (m == 1) {
        const float bd = b_dst[0];
        #pragma unroll
        for (int r = 0; r < 8; ++r) dst[base + r] = acc[r] + bd;
    }
}

// ---------------------------------------------------------------------------
// Phase 2: N x N outer sigmoid / log-prob kernel. Store-bandwidth bound:
// 256 MB output -> ~11 us floor at 23.3 TB/s. One thread per (i,j), one
// 16-byte non-temporal store (output > 192 MB L2, so bypass).
// ---------------------------------------------------------------------------
__global__ __launch_bounds__(256) void pair_log(
    const float* __restrict__ src, const float* __restrict__ dst,
    v4f* __restrict__ out)
{
    const int j = blockIdx.x * 256 + threadIdx.x;  // coalesced lane index
    const int i = blockIdx.y;                      // uniform -> scalar loads

    const float si = src[i];
    const float di = dst[i];
    const float sj = src[j];
    const float dj = dst[j];

    const float a = si + dj;   // logit for s_ij
    const float b = sj + di;   // logit for s_ji

    const float s  = __builtin_amdgcn_rcpf(1.0f + __expf(-a));
    const float t  = __builtin_amdgcn_rcpf(1.0f + __expf(-b));
    const float os = 1.0f - s;
    const float ot = 1.0f - t;

    v4f v;
    v.x = __logf(fmaxf(fminf(os * ot, 1.0f), 1e-10f));  // p_nb
    v.y = __logf(fmaxf(fminf(s  * ot, 1.0f), 1e-10f));  // p_pu
    v.z = __logf(fmaxf(fminf(s  * t , 1.0f), 1e-10f));  // p_pb
    v.w = __logf(fmaxf(fminf(os * t , 1.0f), 1e-10f));  // p_nu

    __builtin_nontemporal_store(v, out + (size_t)i * N_NODES + j);
}

extern "C" void kernel_launch(void* const* d_in, const int* in_sizes, int n_in,
                              void* d_out, int out_size, void* d_ws, size_t ws_size,
                              hipStream_t stream) {
    const float* x     = (const float*)d_in[0];
    const float* w_src = (const float*)d_in[1];
    const float* b_src = (const float*)d_in[2];
    const float* w_dst = (const float*)d_in[3];
    const float* b_dst = (const float*)d_in[4];

    float* ws  = (float*)d_ws;
    float* src = ws;             // [N_NODES]
    float* dst = ws + N_NODES;   // [N_NODES]

    proj_wmma<<<N_NODES / 16, 32, 0, stream>>>(x, w_src, b_src, w_dst, b_dst, src, dst);

    dim3 grid(N_NODES / 256, N_NODES);
    pair_log<<<grid, 256, 0, stream>>>(src, dst, (v4f*)d_out);
}